// GroupedKAAttention_87436944212726
// MI455X (gfx1250) — compile-verified
//
#include <hip/hip_runtime.h>

typedef __attribute__((ext_vector_type(2))) float v2f;
typedef __attribute__((ext_vector_type(8))) float v8f;

namespace {
constexpr int kB  = 4096;   // batch
constexpr int kTD = 4096;   // total dim
constexpr int kG  = 64;     // groups
constexpr int kGS = 64;     // group size
constexpr int kH  = 512;    // hidden
constexpr int kP  = 64;     // patches
constexpr int kE  = 16;     // heads
}

__device__ __forceinline__ v8f wmma4(v2f a, v2f b, v8f c) {
  // D = A(16x4,f32) * B(4x16,f32) + C(16x16,f32), exact fp32 matrix op
  return __builtin_amdgcn_wmma_f32_16x16x4_f32(false, a, false, b, (short)0, c,
                                               false, false);
}

__device__ __forceinline__ float silu(float v) {
  return v / (1.0f + __expf(-v));
}

// ---------------------------------------------------------------------------
// Phase A: per-group SVF MLP for q and k.
// grid = (B/16, GROUPS, 2), block = 256 (8 waves).
// Each wave owns a 64-wide hidden chunk; layer1 (16x16 h-tile, K=64) is fused
// into layer2 via a per-wave LDS transpose; partial f reduced with ds_add_f32.
// Output fbuf layout: [z][b][p][g] (pre-transposed for the global SVF).
// ---------------------------------------------------------------------------
__global__ __launch_bounds__(256) void svf_grouped(
    const float* __restrict__ q, const float* __restrict__ k,
    const float* __restrict__ W1q, const float* __restrict__ b1q,
    const float* __restrict__ W2q, const float* __restrict__ b2q,
    const float* __restrict__ W1k, const float* __restrict__ b1k,
    const float* __restrict__ W2k, const float* __restrict__ b2k,
    float* __restrict__ fbuf) {
  __shared__ float xs[16][68];       // x tile [16 rows][64 k]
  __shared__ float facc[16][64];     // f accumulator [16 rows][64 p]
  __shared__ float stag[8][16][18];  // per-wave h-tile transpose staging

  const int tid  = threadIdx.x;
  const int wave = tid >> 5;
  const int lid  = tid & 31;
  const int mrow = lid & 15;
  const int khi  = (lid >> 4) << 1;  // 0 or 2
  const int hbank = (lid >> 4) << 3; // 0 or 8 (C-frag M offset)

  const int b0 = blockIdx.x * 16;
  const int g  = blockIdx.y;
  const int z  = blockIdx.z;

  const float* __restrict__ x  = z ? k : q;
  const float* __restrict__ W1 = (z ? W1k : W1q) + (size_t)g * kGS * kH;
  const float* __restrict__ b1 = (z ? b1k : b1q) + g * kH;
  const float* __restrict__ W2 = (z ? W2k : W2q) + (size_t)g * kH * kP;
  const float* __restrict__ b2 = (z ? b2k : b2q) + g * kP;

  for (int i = tid; i < 16 * 64; i += 256) {
    const int r = i >> 6, c = i & 63;
    xs[r][c] = x[(size_t)(b0 + r) * kTD + g * kGS + c];
    facc[r][c] = 0.0f;
  }
  __syncthreads();

  v8f fr[4];
#pragma unroll
  for (int pt = 0; pt < 4; ++pt)
#pragma unroll
    for (int e = 0; e < 8; ++e) fr[pt][e] = 0.0f;

  float(*st)[18] = stag[wave];
  const int nwave = wave * 64;

  for (int s = 0; s < 4; ++s) {
    const int nb = nwave + s * 16;  // hidden-column base of this 16-wide slice
    v8f h;
#pragma unroll
    for (int e = 0; e < 8; ++e) h[e] = 0.0f;

    // layer1: h[16x16] = x[16x64] @ W1[:, nb:nb+16]
    const float* w1c = W1 + nb + mrow;  // column n = nb + mrow
#pragma unroll
    for (int kk = 0; kk < kGS; kk += 4) {
      const int k0 = kk + khi;
      v2f a;  a.x = xs[mrow][k0];               a.y = xs[mrow][k0 + 1];
      v2f bf; bf.x = w1c[(size_t)k0 * kH];      bf.y = w1c[(size_t)(k0 + 1) * kH];
      h = wmma4(a, bf, h);
    }

    // bias + silu, then transpose C-layout -> A-layout via wave-private LDS
    const float bias = b1[nb + mrow];
#pragma unroll
    for (int e = 0; e < 8; ++e) st[e + hbank][mrow] = silu(h[e] + bias);

    // layer2 partial: f[16x64] += h[16x16] @ W2[nb:nb+16, :]
#pragma unroll
    for (int kk = 0; kk < 16; kk += 4) {
      const int kh = kk + khi;
      v2f a; a.x = st[mrow][kh]; a.y = st[mrow][kh + 1];
      const float* w2r = W2 + (size_t)(nb + kh) * kP;
#pragma unroll
      for (int pt = 0; pt < 4; ++pt) {
        v2f bf; bf.x = w2r[pt * 16 + mrow]; bf.y = w2r[kP + pt * 16 + mrow];
        fr[pt] = wmma4(a, bf, fr[pt]);
      }
    }
  }

  // cross-wave reduction of f partials
#pragma unroll
  for (int pt = 0; pt < 4; ++pt)
#pragma unroll
    for (int e = 0; e < 8; ++e)
      atomicAdd(&facc[e + hbank][pt * 16 + mrow], fr[pt][e]);
  __syncthreads();

  // store f transposed: fbuf[((z*B + b)*P + p)*G + g]
  for (int i = tid; i < 16 * 64; i += 256) {
    const int r = i >> 6, p = i & 63;
    fbuf[((size_t)(z * kB + b0 + r) * kP + p) * kG + g] = facc[r][p] + b2[p];
  }
}

// ---------------------------------------------------------------------------
// Phase B: global SVF (64 -> 512 -> 16) for q and k + per-row dot -> logits.
// grid = (B), block = 256 (8 waves). Wave w: input = w>>2, p-tile = w&3.
// ---------------------------------------------------------------------------
__global__ __launch_bounds__(256) void svf_global(
    const float* __restrict__ Wg1, const float* __restrict__ bg1,
    const float* __restrict__ Wg2, const float* __restrict__ bg2,
    const float* __restrict__ fbuf, float* __restrict__ logits) {
  __shared__ float ft[2][64][68];    // [inp][p][g] tiles
  __shared__ float stag[8][16][18];  // per-wave transpose staging
  __shared__ float outb[2][64][16];  // [inp][p][head]
  __shared__ float wsum[8];

  const int tid  = threadIdx.x;
  const int wave = tid >> 5;
  const int lid  = tid & 31;
  const int mrow = lid & 15;
  const int khi  = (lid >> 4) << 1;
  const int hbank = (lid >> 4) << 3;
  const int b = blockIdx.x;

  for (int i = tid; i < 4096; i += 256) {
    const int p = i >> 6, gg = i & 63;
    ft[0][p][gg] = fbuf[(size_t)b * 4096 + i];
    ft[1][p][gg] = fbuf[(size_t)(kB + b) * 4096 + i];
  }
  __syncthreads();

  const int inp = wave >> 2;
  const int mt  = wave & 3;
  float(*fp)[68] = ft[inp];
  float(*st)[18] = stag[wave];

  v8f oacc;
#pragma unroll
  for (int e = 0; e < 8; ++e) oacc[e] = 0.0f;

  for (int s = 0; s < 32; ++s) {
    const int nb = s * 16;
    v8f h;
#pragma unroll
    for (int e = 0; e < 8; ++e) h[e] = 0.0f;

    // layer1: h[16x16] = fp[mt*16:+16, 0:64] @ Wg1[:, nb:nb+16]
    const float* w1c = Wg1 + nb + mrow;
#pragma unroll
    for (int kk = 0; kk < 64; kk += 4) {
      const int k0 = kk + khi;
      v2f a;  a.x = fp[mt * 16 + mrow][k0];  a.y = fp[mt * 16 + mrow][k0 + 1];
      v2f bf; bf.x = w1c[(size_t)k0 * kH];   bf.y = w1c[(size_t)(k0 + 1) * kH];
      h = wmma4(a, bf, h);
    }
    const float bias = bg1[nb + mrow];
#pragma unroll
    for (int e = 0; e < 8; ++e) st[e + hbank][mrow] = silu(h[e] + bias);

    // layer2: out[16x16] += h[16x16] @ Wg2[nb:nb+16, 0:16]
#pragma unroll
    for (int kk = 0; kk < 16; kk += 4) {
      const int kh = kk + khi;
      v2f a; a.x = st[mrow][kh]; a.y = st[mrow][kh + 1];
      const float* w2r = Wg2 + (size_t)(nb + kh) * kE;
      v2f bf; bf.x = w2r[mrow]; bf.y = w2r[kE + mrow];
      oacc = wmma4(a, bf, oacc);
    }
  }

  const float be = bg2[mrow];
#pragma unroll
  for (int e = 0; e < 8; ++e)
    outb[inp][mt * 16 + e + hbank][mrow] = oacc[e] + be;
  __syncthreads();

  // attn[b] = sum_{p,e} qo * ko  (1024 products)
  float part = 0.0f;
  const float* qo = &outb[0][0][0];
  const float* ko = &outb[1][0][0];
  for (int i = tid; i < 1024; i += 256) part += qo[i] * ko[i];
#pragma unroll
  for (int off = 16; off; off >>= 1) part += __shfl_xor(part, off, 32);
  if (lid == 0) wsum[wave] = part;
  __syncthreads();
  if (tid == 0) {
    float ssum = 0.0f;
#pragma unroll
    for (int w = 0; w < 8; ++w) ssum += wsum[w];
    logits[b] = ssum;
  }
}

// ---------------------------------------------------------------------------
// Phase C: softmax over the 4096 logits (single block).
// ---------------------------------------------------------------------------
__global__ __launch_bounds__(256) void softmax4096(
    const float* __restrict__ lg, float* __restrict__ out) {
  __shared__ float red[8];
  const int tid = threadIdx.x;
  const int lid = tid & 31;
  const int wv  = tid >> 5;

  float m = -1e30f;
  for (int i = tid; i < kB; i += 256) m = fmaxf(m, lg[i]);
#pragma unroll
  for (int off = 16; off; off >>= 1) m = fmaxf(m, __shfl_xor(m, off, 32));
  if (lid == 0) red[wv] = m;
  __syncthreads();
  float M = red[0];
#pragma unroll
  for (int w = 1; w < 8; ++w) M = fmaxf(M, red[w]);
  __syncthreads();

  float s = 0.0f;
  for (int i = tid; i < kB; i += 256) s += __expf(lg[i] - M);
#pragma unroll
  for (int off = 16; off; off >>= 1) s += __shfl_xor(s, off, 32);
  if (lid == 0) red[wv] = s;
  __syncthreads();
  float S = 0.0f;
#pragma unroll
  for (int w = 0; w < 8; ++w) S += red[w];
  const float inv = 1.0f / S;

  for (int i = tid; i < kB; i += 256) out[i] = __expf(lg[i] - M) * inv;
}

extern "C" void kernel_launch(void* const* d_in, const int* in_sizes, int n_in,
                              void* d_out, int out_size, void* d_ws, size_t ws_size,
                              hipStream_t stream) {
  const float* q   = (const float*)d_in[0];
  const float* k   = (const float*)d_in[1];
  const float* W1q = (const float*)d_in[2];
  const float* b1q = (const float*)d_in[3];
  const float* W2q = (const float*)d_in[4];
  const float* b2q = (const float*)d_in[5];
  const float* W1k = (const float*)d_in[6];
  const float* b1k = (const float*)d_in[7];
  const float* W2k = (const float*)d_in[8];
  const float* b2k = (const float*)d_in[9];
  const float* Wg1 = (const float*)d_in[10];
  const float* bg1 = (const float*)d_in[11];
  const float* Wg2 = (const float*)d_in[12];
  const float* bg2 = (const float*)d_in[13];
  float* out = (float*)d_out;

  float* fbuf   = (float*)d_ws;                      // [2][B][P][G] fp32
  float* logits = fbuf + (size_t)2 * kB * kP * kG;   // [B] fp32

  dim3 gridA(kB / 16, kG, 2);
  svf_grouped<<<gridA, 256, 0, stream>>>(q, k, W1q, b1q, W2q, b2q,
                                         W1k, b1k, W2k, b2k, fbuf);
  svf_global<<<dim3(kB), 256, 0, stream>>>(Wg1, bg1, Wg2, bg2, fbuf, logits);
  softmax4096<<<dim3(1), 256, 0, stream>>>(logits, out);
}